// Transformer_6064493822472
// MI455X (gfx1250) — compile-verified
//
#include <hip/hip_runtime.h>
#include <math.h>

#define LAYERS 3
#define HIDD   256
#define DIN    36
#define DOUTD  128
#define TFR    512
#define NOBJ   3
#define BATCH  4
#define NHEAD  8
#define HD     32
#define TLEN   (TFR * NOBJ)      /* 1536 */
#define MROWS  (BATCH * TLEN)    /* 6144 */
#define EPSV   1e-5f

typedef float v2f __attribute__((ext_vector_type(2)));
typedef float v8f __attribute__((ext_vector_type(8)));

static __device__ __forceinline__ v8f wmma_f32_4(v2f a, v2f b, v8f c) {
  // D = A(16x4) * B(4x16) + C on the fp32 matrix pipe
  return __builtin_amdgcn_wmma_f32_16x16x4_f32(false, a, false, b, (short)0, c,
                                               false, false);
}

// ---------------------------------------------------------------------------
// Register-blocked GEMM: C[M,N] = act(A[M,K] @ W[K,N] + bias (+ Res))
// One wave computes a 32x32 tile (2x2 grid of 16x16 WMMA accumulators):
// per k-step 2 A frags + 2 B frags feed 4 WMMAs. N is a compile-time template
// parameter so the strided B loads (rows k and k+1 of W) collapse to one
// address register + immediate offsets instead of per-step 64-bit adds.
// ACT: 0 = none, 1 = exact GELU
// ---------------------------------------------------------------------------
template <int N, int ACT>
__global__ void gemm32_kernel(const float* __restrict__ A,
                              const float* __restrict__ W,
                              const float* __restrict__ bias,
                              const float* __restrict__ Res,
                              float* __restrict__ C, int K) {
  const int lane = threadIdx.x;
  const int hi = lane >> 4;
  const int lm = lane & 15;
  const int n0 = blockIdx.x * 32;
  const int m0 = blockIdx.y * 32;

  v8f acc00, acc01, acc10, acc11;
  const float bv0 = bias ? bias[n0 + lm] : 0.0f;
  const float bv1 = bias ? bias[n0 + 16 + lm] : 0.0f;
#pragma unroll
  for (int r = 0; r < 8; ++r) {
    acc00[r] = bv0; acc01[r] = bv1;
    acc10[r] = bv0; acc11[r] = bv1;
  }

  const float* arow0 = A + (size_t)(m0 + lm) * K;
  const float* arow1 = A + (size_t)(m0 + 16 + lm) * K;
  const float* wcol = W + (size_t)(2 * hi) * N + n0 + lm;
#pragma unroll 4
  for (int k0 = 0; k0 < K; k0 += 4) {
    v2f a0, a1, b0, b1;
    a0.x = arow0[k0 + 2 * hi];
    a0.y = arow0[k0 + 2 * hi + 1];
    a1.x = arow1[k0 + 2 * hi];
    a1.y = arow1[k0 + 2 * hi + 1];
    const float* wp = wcol + (size_t)k0 * N;
    b0.x = wp[0];
    b0.y = wp[N];
    b1.x = wp[16];
    b1.y = wp[N + 16];
    // Warm caches for the next k-block of W (speculative: OOB is dropped)
    __builtin_prefetch((const void*)(wp + 4 * (size_t)N), 0, 3);
    acc00 = wmma_f32_4(a0, b0, acc00);
    acc01 = wmma_f32_4(a0, b1, acc01);
    acc10 = wmma_f32_4(a1, b0, acc10);
    acc11 = wmma_f32_4(a1, b1, acc11);
  }

#pragma unroll
  for (int r = 0; r < 8; ++r) {
    const int row0 = m0 + r + 8 * hi;
    const int row1 = row0 + 16;
    const int col0 = n0 + lm;
    const int col1 = n0 + 16 + lm;
    float v00 = acc00[r], v01 = acc01[r], v10 = acc10[r], v11 = acc11[r];
    if (Res) {
      v00 += Res[(size_t)row0 * N + col0];
      v01 += Res[(size_t)row0 * N + col1];
      v10 += Res[(size_t)row1 * N + col0];
      v11 += Res[(size_t)row1 * N + col1];
    }
    if (ACT == 1) {
      v00 = 0.5f * v00 * (1.0f + erff(v00 * 0.70710678118654752f));
      v01 = 0.5f * v01 * (1.0f + erff(v01 * 0.70710678118654752f));
      v10 = 0.5f * v10 * (1.0f + erff(v10 * 0.70710678118654752f));
      v11 = 0.5f * v11 * (1.0f + erff(v11 * 0.70710678118654752f));
    }
    C[(size_t)row0 * N + col0] = v00;
    C[(size_t)row0 * N + col1] = v01;
    C[(size_t)row1 * N + col0] = v10;
    C[(size_t)row1 * N + col1] = v11;
  }
}

// ---------------------------------------------------------------------------
// LayerNorm over HIDD per row. blockDim = HIDD.
// ---------------------------------------------------------------------------
__global__ void ln_kernel(const float* __restrict__ X,
                          const float* __restrict__ g,
                          const float* __restrict__ bta,
                          float* __restrict__ Y) {
  __shared__ float sbuf[HIDD];
  const int row = blockIdx.x, tid = threadIdx.x;
  const float v = X[(size_t)row * HIDD + tid];
  sbuf[tid] = v;
  __syncthreads();
  for (int s = HIDD / 2; s > 0; s >>= 1) {
    if (tid < s) sbuf[tid] += sbuf[tid + s];
    __syncthreads();
  }
  const float mean = sbuf[0] * (1.0f / HIDD);
  __syncthreads();
  const float d = v - mean;
  sbuf[tid] = d * d;
  __syncthreads();
  for (int s = HIDD / 2; s > 0; s >>= 1) {
    if (tid < s) sbuf[tid] += sbuf[tid + s];
    __syncthreads();
  }
  const float var = sbuf[0] * (1.0f / HIDD);
  Y[(size_t)row * HIDD + tid] = d * rsqrtf(var + EPSV) * g[tid] + bta[tid];
}

// ---------------------------------------------------------------------------
// BatchNorm (training-mode) channel stats over (B,TF,NO). One block / channel.
// stats[ch] = mean, stats[DIN+ch] = rsqrt(var+eps)
// ---------------------------------------------------------------------------
__global__ void bn_stats_kernel(const float* __restrict__ tok,
                                float* __restrict__ stats) {
  __shared__ float s1[256], s2[256];
  const int ch = blockIdx.x, tid = threadIdx.x;
  float a = 0.0f, b = 0.0f;
  for (int i = tid; i < MROWS; i += 256) {
    const float v = tok[(size_t)i * DIN + ch];
    a += v;
    b += v * v;
  }
  s1[tid] = a;
  s2[tid] = b;
  __syncthreads();
  for (int s = 128; s > 0; s >>= 1) {
    if (tid < s) { s1[tid] += s1[tid + s]; s2[tid] += s2[tid + s]; }
    __syncthreads();
  }
  if (tid == 0) {
    const float mu = s1[0] * (1.0f / MROWS);
    const float var = s2[0] * (1.0f / MROWS) - mu * mu;
    stats[ch] = mu;
    stats[DIN + ch] = rsqrtf(var + EPSV);
  }
}

__global__ void bn_apply_kernel(const float* __restrict__ tok,
                                const float* __restrict__ stats,
                                const float* __restrict__ g,
                                const float* __restrict__ bta,
                                float* __restrict__ xn) {
  const int idx = blockIdx.x * blockDim.x + threadIdx.x;
  if (idx >= MROWS * DIN) return;
  const int ch = idx % DIN;
  xn[idx] = (tok[idx] - stats[ch]) * stats[DIN + ch] * g[ch] + bta[ch];
}

// ---------------------------------------------------------------------------
// Flash attention, one wave per (batch, head, 16-query tile).
// S = Q*K^T via WMMA, online softmax, P*V via WMMA (P re-laid-out via LDS).
// Block-causal frame mask (frame = idx // NOBJ) + key padding mask.
// Q/K/V layout: [b][t][h*HD + d] (i.e. the raw [B,T,HID] projection output).
// ---------------------------------------------------------------------------
__global__ void attn_kernel(const float* __restrict__ Q,
                            const float* __restrict__ Kb,
                            const float* __restrict__ Vb,
                            const float* __restrict__ Mk,
                            float* __restrict__ O) {
  __shared__ float smem[16 * 16];
  const int lane = threadIdx.x;
  const int hi = lane >> 4, lm = lane & 15;
  const int qt = blockIdx.x, hh = blockIdx.y, b = blockIdx.z;
  const int q0 = qt * 16;
  const float scale = 0.1767766952966369f; /* 1/sqrt(32) */

  // Preload Q tile as A fragments for all 8 k-steps (HD=32 / 4)
  v2f qa[8];
  const float* qrow = Q + ((size_t)(b * TLEN + q0 + lm)) * HIDD + hh * HD;
#pragma unroll
  for (int s = 0; s < 8; ++s) {
    qa[s].x = qrow[4 * s + 2 * hi];
    qa[s].y = qrow[4 * s + 2 * hi + 1];
  }

  float rmax[8], rsum[8];
  v8f o0, o1;
#pragma unroll
  for (int r = 0; r < 8; ++r) {
    rmax[r] = -3.0e38f;
    rsum[r] = 0.0f;
    o0[r] = 0.0f;
    o1[r] = 0.0f;
  }

  // Causal truncation: keys only up to the last frame visible by this q-tile
  int kmax = (((q0 + 15) / NOBJ) + 1) * NOBJ;
  if (kmax > TLEN) kmax = TLEN;
  const int nkt = (kmax + 15) / 16;

  for (int kt = 0; kt < nkt; ++kt) {
    const int k0 = kt * 16;
    v8f sAcc;
#pragma unroll
    for (int r = 0; r < 8; ++r) sAcc[r] = 0.0f;

    const float* krow = Kb + ((size_t)(b * TLEN + k0 + lm)) * HIDD + hh * HD;
#pragma unroll
    for (int st = 0; st < 8; ++st) {
      v2f bf;
      bf.x = krow[4 * st + 2 * hi];
      bf.y = krow[4 * st + 2 * hi + 1];
      sAcc = wmma_f32_4(qa[st], bf, sAcc);
    }

    const int key = k0 + lm; // this lane's column (key index)
    const int fk = key / NOBJ;
    const bool keyok = Mk[(size_t)b * TLEN + key] != 0.0f;

    float pr[8];
#pragma unroll
    for (int r = 0; r < 8; ++r) {
      const int qg = q0 + r + 8 * hi;
      float sv = sAcc[r] * scale;
      if (!keyok || (qg / NOBJ) < fk) sv = -3.0e38f;
      float tm = sv;
      tm = fmaxf(tm, __shfl_xor(tm, 1, 32));
      tm = fmaxf(tm, __shfl_xor(tm, 2, 32));
      tm = fmaxf(tm, __shfl_xor(tm, 4, 32));
      tm = fmaxf(tm, __shfl_xor(tm, 8, 32));
      const float nm = fmaxf(rmax[r], tm);
      const float corr = expf(rmax[r] - nm);
      const float p = (sv <= -1.0e37f) ? 0.0f : expf(sv - nm);
      float ts = p;
      ts += __shfl_xor(ts, 1, 32);
      ts += __shfl_xor(ts, 2, 32);
      ts += __shfl_xor(ts, 4, 32);
      ts += __shfl_xor(ts, 8, 32);
      rsum[r] = rsum[r] * corr + ts;
      rmax[r] = nm;
      o0[r] *= corr;
      o1[r] *= corr;
      pr[r] = p;
    }

    // C-layout -> A-layout transpose of P through LDS
    __syncthreads();
#pragma unroll
    for (int r = 0; r < 8; ++r) smem[(r + 8 * hi) * 16 + lm] = pr[r];
    __syncthreads();

    const float* vbase = Vb + ((size_t)(b * TLEN + k0)) * HIDD + hh * HD;
#pragma unroll
    for (int st = 0; st < 4; ++st) {
      const int kkl = 4 * st + 2 * hi;
      v2f pa;
      pa.x = smem[lm * 16 + kkl];
      pa.y = smem[lm * 16 + kkl + 1];
      v2f bv0, bv1;
      bv0.x = vbase[(size_t)kkl * HIDD + lm];
      bv0.y = vbase[(size_t)(kkl + 1) * HIDD + lm];
      bv1.x = vbase[(size_t)kkl * HIDD + 16 + lm];
      bv1.y = vbase[(size_t)(kkl + 1) * HIDD + 16 + lm];
      o0 = wmma_f32_4(pa, bv0, o0);
      o1 = wmma_f32_4(pa, bv1, o1);
    }
  }

#pragma unroll
  for (int r = 0; r < 8; ++r) {
    const float inv = rsum[r] > 0.0f ? 1.0f / rsum[r] : 0.0f;
    float* orow = O + ((size_t)(b * TLEN + q0 + r + 8 * hi)) * HIDD + hh * HD;
    orow[lm] = o0[r] * inv;
    orow[16 + lm] = o1[r] * inv;
  }
}

// ---------------------------------------------------------------------------
// feat_LR: mean over the NOBJ objects per frame.
// ---------------------------------------------------------------------------
__global__ void mean_kernel(const float* __restrict__ Y, float* __restrict__ out) {
  const int idx = blockIdx.x * blockDim.x + threadIdx.x;
  if (idx >= BATCH * TFR * DOUTD) return;
  const int d = idx % DOUTD;
  const int rem = idx / DOUTD;
  const int tf = rem % TFR;
  const int b = rem / TFR;
  const size_t base = ((size_t)(b * TLEN + tf * NOBJ)) * DOUTD + d;
  out[idx] = (Y[base] + Y[base + DOUTD] + Y[base + 2 * DOUTD]) * (1.0f / 3.0f);
}

// ---------------------------------------------------------------------------
extern "C" void kernel_launch(void* const* d_in, const int* in_sizes, int n_in,
                              void* d_out, int out_size, void* d_ws,
                              size_t ws_size, hipStream_t stream) {
  (void)in_sizes; (void)n_in; (void)out_size; (void)ws_size;

  const float* tokens = (const float*)d_in[0];
  const float* masks  = (const float*)d_in[1];
  /* d_in[2] = pos: unused by the reference */
  const float* tok_w  = (const float*)d_in[3];
  const float* tok_b  = (const float*)d_in[4];
  const float* bn_g   = (const float*)d_in[5];
  const float* bn_b   = (const float*)d_in[6];
  const float* ln1_g  = (const float*)d_in[7];
  const float* ln1_b  = (const float*)d_in[8];
  const float* wq     = (const float*)d_in[9];
  const float* bq     = (const float*)d_in[10];
  const float* wk     = (const float*)d_in[11];
  const float* bk     = (const float*)d_in[12];
  const float* wvw    = (const float*)d_in[13];
  const float* bvv    = (const float*)d_in[14];
  const float* wo     = (const float*)d_in[15];
  const float* bo     = (const float*)d_in[16];
  const float* ln2_g  = (const float*)d_in[17];
  const float* ln2_b  = (const float*)d_in[18];
  const float* w1     = (const float*)d_in[19];
  const float* b1     = (const float*)d_in[20];
  const float* w2     = (const float*)d_in[21];
  const float* b2     = (const float*)d_in[22];
  const float* lnf_g  = (const float*)d_in[23];
  const float* lnf_b  = (const float*)d_in[24];
  const float* proj_w = (const float*)d_in[25];
  const float* proj_b = (const float*)d_in[26];

  float* ws = (float*)d_ws;
  const size_t SZ = (size_t)MROWS * HIDD; /* 1.57M floats */
  float* stats = ws;        /* 128 floats: mu[36], rsig[36] */
  float* x  = ws + 128;     /* residual stream  [6144,256] */
  float* h  = x + SZ;       /* LN output        [6144,256] */
  float* q  = h + SZ;       /* Q projection     [6144,256] */
  float* kk = q + SZ;       /* K projection     [6144,256] */
  float* v  = kk + SZ;      /* V projection     [6144,256] */
  float* o  = v + SZ;       /* attn out / xn / final proj  */
  float* m1 = q;            /* MLP hidden [6144,1024] aliases q..o */

  const dim3 wv32(32);
  const dim3 gHID(HIDD / 32, MROWS / 32);

  // --- BatchNorm + token embedding ---
  bn_stats_kernel<<<DIN, 256, 0, stream>>>(tokens, stats);
  bn_apply_kernel<<<(MROWS * DIN + 255) / 256, 256, 0, stream>>>(
      tokens, stats, bn_g, bn_b, o);
  gemm32_kernel<HIDD, 0><<<gHID, wv32, 0, stream>>>(
      o, tok_w, tok_b, nullptr, x, DIN);

  // --- Transformer layers ---
  for (int l = 0; l < LAYERS; ++l) {
    const size_t wO = (size_t)l * HIDD * HIDD;
    ln_kernel<<<MROWS, HIDD, 0, stream>>>(x, ln1_g + l * HIDD, ln1_b + l * HIDD, h);
    gemm32_kernel<HIDD, 0><<<gHID, wv32, 0, stream>>>(
        h, wq + wO, bq + l * HIDD, nullptr, q, HIDD);
    gemm32_kernel<HIDD, 0><<<gHID, wv32, 0, stream>>>(
        h, wk + wO, bk + l * HIDD, nullptr, kk, HIDD);
    gemm32_kernel<HIDD, 0><<<gHID, wv32, 0, stream>>>(
        h, wvw + wO, bvv + l * HIDD, nullptr, v, HIDD);
    attn_kernel<<<dim3(TLEN / 16, NHEAD, BATCH), wv32, 0, stream>>>(
        q, kk, v, masks, o);
    gemm32_kernel<HIDD, 0><<<gHID, wv32, 0, stream>>>(
        o, wo + wO, bo + l * HIDD, x, x, HIDD);

    ln_kernel<<<MROWS, HIDD, 0, stream>>>(x, ln2_g + l * HIDD, ln2_b + l * HIDD, h);
    gemm32_kernel<4 * HIDD, 1><<<dim3(4 * HIDD / 32, MROWS / 32), wv32, 0, stream>>>(
        h, w1 + (size_t)l * HIDD * 4 * HIDD, b1 + l * 4 * HIDD, nullptr, m1, HIDD);
    gemm32_kernel<HIDD, 0><<<gHID, wv32, 0, stream>>>(
        m1, w2 + (size_t)l * 4 * HIDD * HIDD, b2 + l * HIDD, x, x, 4 * HIDD);
  }

  // --- Final LN + projection + mean over objects ---
  ln_kernel<<<MROWS, HIDD, 0, stream>>>(x, lnf_g, lnf_b, h);
  gemm32_kernel<DOUTD, 0><<<dim3(DOUTD / 32, MROWS / 32), wv32, 0, stream>>>(
      h, proj_w, proj_b, nullptr, o, HIDD);
  mean_kernel<<<(BATCH * TFR * DOUTD + 255) / 256, 256, 0, stream>>>(
      o, (float*)d_out);
}